// LevelSelect_30502857736595
// MI455X (gfx1250) — compile-verified
//
#include <hip/hip_runtime.h>
#include <hip/hip_bf16.h>

// ---- problem constants (match reference) ----
#define NB        8
#define TLOC      13343          // 10000+2500+625+169+49
#define NCLS      80
#define NGT       100
#define EPSC      1e-6f
#define ALPHA_C   0.25f
#define ONE_M_A   0.75f

typedef float v2f  __attribute__((ext_vector_type(2)));
typedef float v8f  __attribute__((ext_vector_type(8)));
typedef unsigned int u32x4 __attribute__((ext_vector_type(4)));
typedef int   i32x4 __attribute__((ext_vector_type(4)));
typedef int   i32x8 __attribute__((ext_vector_type(8)));

#if defined(__has_builtin)
#  if __has_builtin(__builtin_amdgcn_tensor_load_to_lds) && \
      __has_builtin(__builtin_amdgcn_s_wait_tensorcnt)
#    define HAVE_TDM 1
#  endif
#endif
#if __has_include(<hip/amd_detail/amd_gfx1250_TDM.h>)
#  define TDM_6ARG 1   // amdgpu-toolchain (clang-23): 6-arg builtin
#endif

// generic -> LDS byte offset (addrspacecast + ptrtoint)
__device__ __forceinline__ unsigned lds_byte_offset(const void* p) {
    return (unsigned)(uintptr_t)(__attribute__((address_space(3))) const void*)p;
}

// =====================================================================
// Kernel 1: s_neg[b*T+t] = sum_c 0.75 * p^2 * (-log(1-p)) via WMMA.
// One wave = one 16-row tile. TDM stages the tile's 16x80 f32 block
// (5120 B, contiguous) into this wave's private LDS slice with a single
// tensor_load_to_lds; the wave waits once on TENSORcnt, then runs the
// focal/log math + 20 chained V_WMMA_F32_16X16X4_F32 (A = focal-neg
// terms, B = ones => D rows are the 80-class sums). No barrier needed:
// each wave consumes only its own slice. OOB rows of the last tile are
// clipped by tensor_dim0 (TDM returns zeros past the bound).
// =====================================================================
__global__ void __launch_bounds__(256)
sneg_wmma_kernel(const float* __restrict__ cls, float* __restrict__ sneg, int total)
{
    const int lane  = threadIdx.x & 31;
    const int wave  = threadIdx.x >> 5;
    const int tile  = blockIdx.x * 8 + wave;
    const int t0    = tile * 16;
    const int M     = lane & 15;       // row within 16x4 A fragment
    const int khalf = lane >> 4;       // lanes 16..31 hold K=2,3

#if defined(HAVE_TDM)
    __shared__ float lds[8 * 1280];              // 8 waves x 16 rows x 80 f32 = 40 KB
    float* slice = &lds[wave * 1280];

    // ---- build Tensor DMA Descriptor (D#), CDNA5 ISA ----
    const unsigned long long ga =
        (unsigned long long)(uintptr_t)(cls + (size_t)t0 * NCLS);
    const unsigned ldsOff = lds_byte_offset(slice);
    const unsigned remElems =
        (unsigned)min(16 * NCLS, (total - t0) * NCLS);   // OOB clip for last tile

    u32x4 g0;
    g0[0] = 1u;                                   // count=1, no gather
    g0[1] = ldsOff;                               // lds_addr (bytes)
    g0[2] = (unsigned)(ga & 0xFFFFFFFFull);       // global_addr[31:0]
    g0[3] = (unsigned)((ga >> 32) & 0x1FFFFFFull) // global_addr[56:32]
          | 0x80000000u;                          // type=2 ("image")

    i32x8 g1;
    g1[0] = (int)(2u << 16);                      // workgroup_mask=0, data_size=4B
    g1[1] = (int)(remElems << 16);                // tensor_dim0[15:0] in [31:16]
    g1[2] = (int)(remElems >> 16);                // tensor_dim0[31:16]; tensor_dim1=0
    g1[3] = (int)(1280u << 16);                   // tile_dim0 = 1280 in [127:112]
    g1[4] = 0;                                    // tile_dim1/2 unused
    g1[5] = 1280;                                 // tensor_dim0_stride (benign)
    g1[6] = 0;
    g1[7] = 0;

    i32x4 gz = {0, 0, 0, 0};                      // groups 2/3: <=2D tensor

#if defined(TDM_6ARG)
    i32x8 gz8 = {0, 0, 0, 0, 0, 0, 0, 0};
    __builtin_amdgcn_tensor_load_to_lds(g0, g1, gz, gz, gz8, 0);
#else
    __builtin_amdgcn_tensor_load_to_lds(g0, g1, gz, gz, 0);
#endif
    __builtin_amdgcn_s_wait_tensorcnt(0);         // TDM transfer complete

    // per-lane float2 view inside this wave's slice: element M*80 + 4k + 2*khalf
    const float2* __restrict__ row =
        reinterpret_cast<const float2*>(slice + M * NCLS) + khalf;
#else
    int bt = t0 + M;
    if (bt >= total) bt = total - 1;              // clamp pad rows (loads only)
    const float2* __restrict__ row =
        reinterpret_cast<const float2*>(cls) + (size_t)bt * 40 + khalf;
#endif

    // ---- focal-neg terms + WMMA row-sum accumulation ----
    v8f c = {0.f, 0.f, 0.f, 0.f, 0.f, 0.f, 0.f, 0.f};
    v2f ones; ones.x = 1.0f; ones.y = 1.0f;       // B = all ones -> row sums

#pragma unroll
    for (int k = 0; k < 20; ++k) {
        float2 v = row[2 * k];                    // classes (4k+2*khalf, +1)
        float p0 = fminf(fmaxf(v.x, EPSC), 1.0f - EPSC);
        float p1 = fminf(fmaxf(v.y, EPSC), 1.0f - EPSC);
        v2f a;
        a.x = ONE_M_A * p0 * p0 * (-__logf(1.0f - p0));
        a.y = ONE_M_A * p1 * p1 * (-__logf(1.0f - p1));
        // D = A x B + C  (16x16x4 f32 WMMA)
        c = __builtin_amdgcn_wmma_f32_16x16x4_f32(
                false, a, false, ones, (short)0, c, false, false);
    }

    // D layout: VGPR r -> M=r (lanes 0-15) / M=r+8 (lanes 16-31); every
    // column is identical (B==1): lane 0 holds rows 0..7, lane 16 rows 8..15.
    if ((lane & 15) == 0) {
        const int base = t0 + (khalf ? 8 : 0);
#pragma unroll
        for (int r = 0; r < 8; ++r) {
            int t = base + r;
            if (t < total) sneg[t] = c[r];
        }
    }
}

// =====================================================================
// Kernel 2: one wave per GT box. For each of the 5 FPN levels, accumulate
// masked cls + IoU loss over the center region, wave-reduce, argmin, store.
// =====================================================================
__global__ void __launch_bounds__(32)
level_select_kernel(const float* __restrict__ cls,
                    const float* __restrict__ regr,
                    const float* __restrict__ gt,
                    const float* __restrict__ sneg,
                    int* __restrict__ out)
{
    const int box  = blockIdx.x;          // 0..799
    const int b    = box / NGT;
    const int lane = threadIdx.x;

    const float* g = gt + (size_t)box * 5;
    const float b0 = g[0], b1 = g[1], b2 = g[2], b3 = g[3];
    const int   label = (int)g[4];

    if (fabsf(b0) + fabsf(b1) + fabsf(b2) + fabsf(b3) <= 0.0f) {
        if (lane == 0) out[box] = -1;
        return;
    }

    const size_t batchBase = (size_t)b * TLOC;
    float bestLoss = 3.4e38f;
    int   bestLvl  = 0;

#pragma unroll
    for (int lvl = 0; lvl < 5; ++lvl) {
        const int fdim   = (lvl == 0) ? 100 : (lvl == 1) ? 50 : (lvl == 2) ? 25 : (lvl == 3) ? 13 : 7;
        const int stride = 8 << lvl;
        const int start  = (lvl == 0) ? 0 : (lvl == 1) ? 10000 : (lvl == 2) ? 12500
                         : (lvl == 3) ? 13125 : 13294;
        const float fs  = (float)stride;
        const float inv = 1.0f / fs;

        const float pb0 = b0 * inv, pb1 = b1 * inv, pb2 = b2 * inv, pb3 = b3 * inv;
        const float cx = (pb0 + pb2) * 0.5f, cy = (pb1 + pb3) * 0.5f;
        const float hw = (pb2 - pb0) * 0.1f;     // POS_SCALE/2 = 0.1
        const float hh = (pb3 - pb1) * 0.1f;

        int x1 = (int)fminf(fmaxf(floorf(cx - hw), 0.0f), (float)(fdim - 1));
        int y1 = (int)fminf(fmaxf(floorf(cy - hh), 0.0f), (float)(fdim - 1));
        int x2 = (int)ceilf(cx + hw); x2 = min(max(x2, x1 + 1), fdim);
        int y2 = (int)ceilf(cy + hh); y2 = min(max(y2, y1 + 1), fdim);

        const int w = x2 - x1;
        const int n = w * (y2 - y1);

        // hint: pull the s_neg slab for this level toward the caches
        __builtin_prefetch((const void*)(sneg + batchBase + start + (size_t)y1 * fdim + x1), 0, 0);

        float acc = 0.0f;
        for (int i = lane; i < n; i += 32) {
            const int yy = y1 + i / w;
            const int xx = x1 + i % w;
            const size_t bt = batchBase + (size_t)start + (size_t)yy * fdim + xx;

            // classification part
            const float s = sneg[bt];
            float p = cls[bt * (size_t)NCLS + label];
            p = fminf(fmaxf(p, EPSC), 1.0f - EPSC);
            const float omp     = 1.0f - p;
            const float neg_lab = ONE_M_A * p * p   * (-__logf(omp));
            const float pos_lab = ALPHA_C * omp * omp * (-__logf(p));
            const float cls_loc = s - neg_lab + pos_lab;

            // IoU part
            const float4 rp = reinterpret_cast<const float4*>(regr)[bt];
            const float sx = ((float)xx + 0.5f) * fs;
            const float sy = ((float)yy + 0.5f) * fs;
            const float tl = (sx - b0) * 0.25f;
            const float tt = (sy - b1) * 0.25f;
            const float tr = (b2 - sx) * 0.25f;
            const float tb = (b3 - sy) * 0.25f;
            const float t_area = (tl + tr) * (tt + tb);
            const float p_area = (rp.x + rp.z) * (rp.y + rp.w);
            const float wi = fminf(rp.x, tl) + fminf(rp.z, tr);
            const float hi = fminf(rp.y, tt) + fminf(rp.w, tb);
            const float ai = wi * hi;
            const float un = t_area + p_area - ai;
            const float iou_loc = -__logf((ai + 1.0f) / (un + 1.0f));

            acc += cls_loc + iou_loc;
        }

        // wave32 butterfly reduction -> every lane has the full sum
        acc += __shfl_xor(acc, 16, 32);
        acc += __shfl_xor(acc, 8, 32);
        acc += __shfl_xor(acc, 4, 32);
        acc += __shfl_xor(acc, 2, 32);
        acc += __shfl_xor(acc, 1, 32);

        const float loss = acc / (float)max(n, 1);
        if (loss < bestLoss) { bestLoss = loss; bestLvl = lvl; }
    }

    if (lane == 0) out[box] = bestLvl;
}

extern "C" void kernel_launch(void* const* d_in, const int* in_sizes, int n_in,
                              void* d_out, int out_size, void* d_ws, size_t ws_size,
                              hipStream_t stream)
{
    const float* cls  = (const float*)d_in[0];   // (8, 13343, 80) f32
    const float* regr = (const float*)d_in[1];   // (8, 13343, 4)  f32
    // d_in[2] = feature_shapes (compile-time constants here)
    const float* gt   = (const float*)d_in[3];   // (8, 100, 5)    f32

    float* sneg = (float*)d_ws;                  // 106744 floats = 427 KB scratch
    const int total = NB * TLOC;                 // 106744

    // 106744 rows -> 6672 tiles of 16 -> 834 blocks x 8 waves exactly
    sneg_wmma_kernel<<<834, 256, 0, stream>>>(cls, sneg, total);

    // one wave per (batch, gt-box)
    level_select_kernel<<<NB * NGT, 32, 0, stream>>>(cls, regr, gt, sneg, (int*)d_out);
}